// GCN_61830349193252
// MI455X (gfx1250) — compile-verified
//
#include <hip/hip_runtime.h>
#include <hip/hip_bf16.h>

#define N_NODES 100000
#define N_EDGES 1600000
#define F_IN    128
#define F_HID   64
#define F_OUT   40
#define BN_EPS  1e-5f

typedef __attribute__((ext_vector_type(2))) float v2f;
typedef __attribute__((ext_vector_type(8))) float v8f;

__device__ __forceinline__ void atomAddF(float* p, float v) {
  __hip_atomic_fetch_add(p, v, __ATOMIC_RELAXED, __HIP_MEMORY_SCOPE_AGENT);
}

// ---------------------------------------------------------------- utilities
__global__ void fill_zero_kernel(float* __restrict__ p, long long n) {
  long long t = (long long)blockIdx.x * blockDim.x + threadIdx.x;
  if (t < n) p[t] = 0.0f;
}

__global__ void degree_kernel(const long long* __restrict__ dst,
                              float* __restrict__ deg) {
  int e = blockIdx.x * blockDim.x + threadIdx.x;
  if (e < N_EDGES) atomAddF(&deg[(int)dst[e]], 1.0f);
}

__global__ void dinv_kernel(float* __restrict__ deg_to_dinv) {
  int n = blockIdx.x * blockDim.x + threadIdx.x;
  if (n < N_NODES) deg_to_dinv[n] = rsqrtf(deg_to_dinv[n] + 1.0f);  // +1 self loop
}

// ---------------------------------------------------------------- GEMM 1
// H[N_NODES, 64] = X[N_NODES, 128] @ W[128, 64], fp32 WMMA 16x16x4.
// One wave per 16x16 output tile. 100000 % 16 == 0 -> no M tail.
__global__ void gemm1_wmma_kernel(const float* __restrict__ X,
                                  const float* __restrict__ W,
                                  float* __restrict__ H) {
  const int wave = (blockIdx.x * blockDim.x + threadIdx.x) >> 5;
  const int lane = threadIdx.x & 31;
  const int NT_N = F_HID / 16;                 // 4 tiles along N
  const int m0 = (wave / NT_N) * 16;
  const int n0 = (wave % NT_N) * 16;
  if (m0 >= N_NODES) return;                   // wave-uniform exit (EXEC all-1s for WMMA)

  const int half = lane >> 4;                  // 0: K pair {0,1}, 1: K pair {2,3}
  const int l15  = lane & 15;

  v8f acc = {};
#pragma unroll 4
  for (int k0 = 0; k0 < F_IN; k0 += 4) {
    const int kb = k0 + half * 2;
    // A 16x4: lane l15 = row m, VGPR0/1 = K kb/kb+1
    v2f a, b;
    a.x = X[(size_t)(m0 + l15) * F_IN + kb];
    a.y = X[(size_t)(m0 + l15) * F_IN + kb + 1];
    // B 4x16: lane l15 = col n, VGPR0/1 = K rows kb/kb+1
    b.x = W[(size_t)kb       * F_HID + n0 + l15];
    b.y = W[(size_t)(kb + 1) * F_HID + n0 + l15];
    acc = __builtin_amdgcn_wmma_f32_16x16x4_f32(false, a, false, b,
                                                (short)0, acc, false, false);
  }
  // C/D: VGPR r -> rows r (lanes 0-15) and r+8 (lanes 16-31), col = l15
#pragma unroll
  for (int r = 0; r < 8; ++r) {
    H[(size_t)(m0 + r + half * 8) * F_HID + n0 + l15] = acc[r];
  }
}

// ---------------------------------------------------------------- GEMM 2
// H2[N_NODES, 40] = HB[N_NODES, 64] @ W2[64, 40]; N padded to 3 tiles of 16.
__global__ void gemm2_wmma_kernel(const float* __restrict__ HB,
                                  const float* __restrict__ W2,
                                  float* __restrict__ H2) {
  const int wave = (blockIdx.x * blockDim.x + threadIdx.x) >> 5;
  const int lane = threadIdx.x & 31;
  const int NT_N = 3;                          // ceil(40/16)
  const int m0 = (wave / NT_N) * 16;
  const int n0 = (wave % NT_N) * 16;
  if (m0 >= N_NODES) return;                   // wave-uniform

  const int half = lane >> 4;
  const int l15  = lane & 15;
  const int col  = n0 + l15;
  const bool cok = (col < F_OUT);

  v8f acc = {};
#pragma unroll 4
  for (int k0 = 0; k0 < F_HID; k0 += 4) {
    const int kb = k0 + half * 2;
    v2f a, b;
    a.x = HB[(size_t)(m0 + l15) * F_HID + kb];
    a.y = HB[(size_t)(m0 + l15) * F_HID + kb + 1];
    b.x = cok ? W2[(size_t)kb       * F_OUT + col] : 0.0f;
    b.y = cok ? W2[(size_t)(kb + 1) * F_OUT + col] : 0.0f;
    acc = __builtin_amdgcn_wmma_f32_16x16x4_f32(false, a, false, b,
                                                (short)0, acc, false, false);
  }
  if (cok) {
#pragma unroll
    for (int r = 0; r < 8; ++r) {
      H2[(size_t)(m0 + r + half * 8) * F_OUT + col] = acc[r];
    }
  }
}

// ---------------------------------------------------------------- edge scatter, layer 1 (F=64)
// flat thread = edge * 16 + chunk; each thread: float4 gather + 4 fp32 atomics.
__global__ void scatter1_kernel(const long long* __restrict__ src,
                                const long long* __restrict__ dst,
                                const float* __restrict__ dinv,
                                const float* __restrict__ h,
                                float* __restrict__ agg) {
  long long t = (long long)blockIdx.x * blockDim.x + threadIdx.x;
  if (t >= (long long)N_EDGES * 16) return;
  const int e = (int)(t >> 4);
  const int c = ((int)t & 15) * 4;
  const int s = (int)src[e];
  const int d = (int)dst[e];
  const float w = dinv[s] * dinv[d];
  const float4 v = *(const float4*)(h + (size_t)s * F_HID + c);
  float* p = agg + (size_t)d * F_HID + c;
  atomAddF(p + 0, v.x * w);
  atomAddF(p + 1, v.y * w);
  atomAddF(p + 2, v.z * w);
  atomAddF(p + 3, v.w * w);
}

// ---------------------------------------------------------------- edge scatter, layer 2 (F=40)
__global__ void scatter2_kernel(const long long* __restrict__ src,
                                const long long* __restrict__ dst,
                                const float* __restrict__ dinv,
                                const float* __restrict__ h2,
                                float* __restrict__ out) {
  long long t = (long long)blockIdx.x * blockDim.x + threadIdx.x;
  if (t >= (long long)N_EDGES * 10) return;
  const int e = (int)(t / 10);
  const int c = ((int)(t % 10)) * 4;
  const int s = (int)src[e];
  const int d = (int)dst[e];
  const float w = dinv[s] * dinv[d];
  const float4 v = *(const float4*)(h2 + (size_t)s * F_OUT + c);
  float* p = out + (size_t)d * F_OUT + c;
  atomAddF(p + 0, v.x * w);
  atomAddF(p + 1, v.y * w);
  atomAddF(p + 2, v.z * w);
  atomAddF(p + 3, v.w * w);
}

// ---------------------------------------------------------------- self-loop + bias + BN + ReLU
// hb[n,c] = relu(BN(agg[n,c] + h1[n,c]*dinv[n]^2 + b1[c]))   (written in place over agg)
__global__ void bn_relu_kernel(const float* __restrict__ h1,
                               const float* __restrict__ dinv,
                               const float* __restrict__ b1,
                               const float* __restrict__ gamma,
                               const float* __restrict__ beta,
                               const float* __restrict__ mean,
                               const float* __restrict__ var,
                               float* __restrict__ agg) {
  long long t = (long long)blockIdx.x * blockDim.x + threadIdx.x;
  if (t >= (long long)N_NODES * F_HID) return;
  const int n = (int)(t >> 6);      // /64
  const int c = (int)t & 63;
  const float di = dinv[n];
  float v = agg[t] + h1[t] * (di * di) + b1[c];
  v = (v - mean[c]) * rsqrtf(var[c] + BN_EPS) * gamma[c] + beta[c];
  agg[t] = fmaxf(v, 0.0f);
}

// ---------------------------------------------------------------- layer-2 self-loop + bias
__global__ void finalize_kernel(const float* __restrict__ h2,
                                const float* __restrict__ dinv,
                                const float* __restrict__ b2,
                                float* __restrict__ out) {
  long long t = (long long)blockIdx.x * blockDim.x + threadIdx.x;
  if (t >= (long long)N_NODES * F_OUT) return;
  const int n = (int)(t / F_OUT);
  const int c = (int)(t % F_OUT);
  const float di = dinv[n];
  out[t] += h2[t] * (di * di) + b2[c];
}

// ---------------------------------------------------------------- driver
extern "C" void kernel_launch(void* const* d_in, const int* in_sizes, int n_in,
                              void* d_out, int out_size, void* d_ws, size_t ws_size,
                              hipStream_t stream) {
  const float*     x    = (const float*)d_in[0];
  const long long* ei   = (const long long*)d_in[1];   // [2, E] int64
  const float*     W1   = (const float*)d_in[2];
  const float*     b1   = (const float*)d_in[3];
  const float*     gmm  = (const float*)d_in[4];
  const float*     beta = (const float*)d_in[5];
  const float*     rmu  = (const float*)d_in[6];
  const float*     rvar = (const float*)d_in[7];
  const float*     W2   = (const float*)d_in[8];
  const float*     b2   = (const float*)d_in[9];
  float* out = (float*)d_out;

  const long long* src = ei;             // row 0
  const long long* dst = ei + N_EDGES;   // row 1

  // workspace layout (floats)
  float* ws   = (float*)d_ws;
  float* dinv = ws;                                   // N        (deg -> dinv in place)
  float* h1   = dinv + N_NODES;                       // N*64
  float* agg  = h1 + (size_t)N_NODES * F_HID;         // N*64
  float* h2   = agg + (size_t)N_NODES * F_HID;        // N*40

  const int B = 256;
  auto blocks = [](long long n, int b) { return (unsigned)((n + b - 1) / b); };

  // zero: degree, layer-1 accumulator, output accumulator
  fill_zero_kernel<<<blocks(N_NODES, B), B, 0, stream>>>(dinv, N_NODES);
  fill_zero_kernel<<<blocks((long long)N_NODES * F_HID, B), B, 0, stream>>>(agg, (long long)N_NODES * F_HID);
  fill_zero_kernel<<<blocks((long long)N_NODES * F_OUT, B), B, 0, stream>>>(out, (long long)N_NODES * F_OUT);

  // degrees -> dinv
  degree_kernel<<<blocks(N_EDGES, B), B, 0, stream>>>(dst, dinv);
  dinv_kernel<<<blocks(N_NODES, B), B, 0, stream>>>(dinv);

  // layer 1: GEMM (WMMA fp32), edge scatter, BN+ReLU
  {
    const int waves = (N_NODES / 16) * (F_HID / 16);          // 25000
    gemm1_wmma_kernel<<<blocks((long long)waves * 32, 128), 128, 0, stream>>>(x, W1, h1);
  }
  scatter1_kernel<<<blocks((long long)N_EDGES * 16, B), B, 0, stream>>>(src, dst, dinv, h1, agg);
  bn_relu_kernel<<<blocks((long long)N_NODES * F_HID, B), B, 0, stream>>>(h1, dinv, b1, gmm, beta, rmu, rvar, agg);

  // layer 2: GEMM (WMMA fp32), edge scatter, self-loop + bias
  {
    const int waves = (N_NODES / 16) * 3;                     // 18750
    gemm2_wmma_kernel<<<blocks((long long)waves * 32, 128), 128, 0, stream>>>(agg, W2, h2);
  }
  scatter2_kernel<<<blocks((long long)N_EDGES * 10, B), B, 0, stream>>>(src, dst, dinv, h2, out);
  finalize_kernel<<<blocks((long long)N_NODES * F_OUT, B), B, 0, stream>>>(h2, dinv, b2, out);
}